// MultiModalPromptLearner_65566970741163
// MI455X (gfx1250) — compile-verified
//
#include <hip/hip_runtime.h>
#include <hip/hip_bf16.h>

typedef __attribute__((ext_vector_type(2))) float v2f;
typedef __attribute__((ext_vector_type(4))) float v4f;
typedef __attribute__((ext_vector_type(8))) float v8f;

#define B_    4096
#define L_    77
#define D_    512
#define NCTX_ 4
#define DEPTH_ 9
#define VD_   768

// Output tuple layout (flat float offsets, in return order):
// pe (B,L,D), pt (B,L), ctx (NCTX,D), proj_ctx (NCTX,VD), cpt (8,NCTX,D), vdp (8,NCTX,VD)
constexpr long long OFF_PE   = 0;
constexpr long long OFF_PT   = (long long)B_ * L_ * D_;              // 161,480,704
constexpr long long OFF_CTX  = OFF_PT  + (long long)B_ * L_;         // +315,392
constexpr long long OFF_PROJ = OFF_CTX + (long long)NCTX_ * D_;      // +2,048
constexpr long long OFF_CPT  = OFF_PROJ + (long long)NCTX_ * VD_;    // +3,072
constexpr long long OFF_VDP  = OFF_CPT + (long long)(DEPTH_-1) * NCTX_ * D_; // +16,384

// ---------------------------------------------------------------------------
// Kernel 1: build prompt tokens pt (one wave32 per batch row).
// argmax (first occurrence) over 77 int32 tokens via butterfly shuffle;
// eot = max value itself (take_along_axis at argmax).
// ---------------------------------------------------------------------------
__global__ __launch_bounds__(32) void build_pt_kernel(
    const int* __restrict__ text, float* __restrict__ out, int* __restrict__ tok_ws) {
  const int b = blockIdx.x;
  const int lane = threadIdx.x;
  const int* __restrict__ row = text + (long long)b * L_;

  // per-lane first-occurrence max over strided elements (indices increase per lane)
  int bestV = (int)0x80000000;
  int bestI = 0;
  for (int l = lane; l < L_; l += 32) {
    int v = row[l];
    if (v > bestV) { bestV = v; bestI = l; }
  }
  // wave32 butterfly reduction: max value, tie -> smaller index
  #pragma unroll
  for (int off = 16; off >= 1; off >>= 1) {
    int ov = __shfl_xor(bestV, off, 32);
    int oi = __shfl_xor(bestI, off, 32);
    if (ov > bestV || (ov == bestV && oi < bestI)) { bestV = ov; bestI = oi; }
  }
  const int eot = bestV;  // text[b, argmax]

  for (int l = lane; l < L_; l += 32) {
    int t;
    if (l == 0)            t = row[0];
    else if (l <= NCTX_)   t = 0;
    else                   t = row[l - NCTX_];   // pt[5..76] = text[1..72]
    if (l == L_ - 1 && t != 0) t = eot;          // last position -> eot if nonzero
    tok_ws[(long long)b * L_ + l] = t;
    out[OFF_PT + (long long)b * L_ + l] = (float)t;
  }
}

// ---------------------------------------------------------------------------
// Kernel 2: pe gather (the 647 MB stream). One float4 per thread.
// emb (101 MB) stays hot in the 192 MB L2; pe writes are non-temporal so the
// streaming output does not evict it.
// ---------------------------------------------------------------------------
__global__ __launch_bounds__(256) void gather_pe_kernel(
    const float* __restrict__ emb, const float* __restrict__ ctx,
    const int* __restrict__ tok, float* __restrict__ pe_out) {
  const long long total = (long long)B_ * L_ * (D_ / 4);
  long long e = (long long)blockIdx.x * blockDim.x + threadIdx.x;
  if (e >= total) return;
  const int row = (int)(e >> 7);      // D_/4 == 128 float4 per row
  const int col = (int)(e & 127);
  const int l = row % L_;

  const float* __restrict__ src;
  if (l >= 1 && l <= NCTX_) {
    src = ctx + (long long)(l - 1) * D_ + col * 4;
  } else {
    const int t = tok[row];
    src = emb + (long long)t * D_ + col * 4;
  }
  v4f v = *(const v4f*)src;
  __builtin_nontemporal_store(v, (v4f*)(pe_out + (long long)row * D_ + col * 4));
}

// ---------------------------------------------------------------------------
// Kernel 3: pass-through copies of ctx and cpt into the output tuple.
// ---------------------------------------------------------------------------
__global__ __launch_bounds__(256) void copy_small_kernel(
    const float* __restrict__ ctx, const float* __restrict__ cpt, float* __restrict__ out) {
  const int i = blockIdx.x * blockDim.x + threadIdx.x;
  if (i < NCTX_ * D_)               out[OFF_CTX + i] = ctx[i];
  if (i < (DEPTH_-1) * NCTX_ * D_)  out[OFF_CPT + i] = cpt[i];
}

// ---------------------------------------------------------------------------
// Kernel 4: tiny GEMMs via v_wmma_f32_16x16x4_f32 (full f32 precision).
//   slice 0      : proj_ctx = ctx @ proj_w^T + proj_b          (M=4,N=768,K=512)
//   slice 1..8=k : vdp[k]   = cpt[k] @ cw[k]^T + cb[k]         (M=4,N=768,K=512)
// One wave per 16-wide N tile: 9 slices * 48 tiles = 432 waves.
// Fragment layouts per CDNA5 ISA 7.12.2 (32-bit A 16x4, B 4x16, C 16x16).
// M=4 padded to 16 branch-free: A row address clamped to (m & 3), fragment
// multiplied by a 0/1 lane mask -> no EXEC manipulation in the WMMA loop.
// ---------------------------------------------------------------------------
__global__ __launch_bounds__(32) void wmma_proj_kernel(
    const float* __restrict__ ctx, const float* __restrict__ proj_w,
    const float* __restrict__ proj_b, const float* __restrict__ cpt,
    const float* __restrict__ cw, const float* __restrict__ cb,
    float* __restrict__ out) {
  const int bx    = blockIdx.x;
  const int tile  = bx % (VD_ / 16);
  const int slice = bx / (VD_ / 16);

  const float* A; const float* W; const float* bias; long long obase;
  if (slice == 0) {
    A = ctx; W = proj_w; bias = proj_b; obase = OFF_PROJ;
  } else {
    const int k = slice - 1;
    A    = cpt + (long long)k * NCTX_ * D_;
    W    = cw  + (long long)k * VD_ * D_;
    bias = cb  + (long long)k * VD_;
    obase = OFF_VDP + (long long)k * NCTX_ * VD_;
  }

  const int lane = threadIdx.x;          // wave32
  const int m  = lane & 15;              // A row / B column within tile
  const int kg = (lane >> 4) << 1;       // K group: lanes 16-31 hold K+2
  const int n0 = tile * 16;

  // clamp row so every lane has an in-bounds address; mask zeroes rows 4..15
  const float amask = (m < NCTX_) ? 1.0f : 0.0f;
  const float* __restrict__ arow = A + (long long)(m & (NCTX_ - 1)) * D_ + kg;
  const float* __restrict__ wrow = W + (long long)(n0 + m) * D_ + kg;  // B[k][n] = W[n][k]

  v8f acc = {};
  for (int kb = 0; kb < D_; kb += 4) {
    v2f a = *(const v2f*)(arow + kb);
    a *= amask;                          // branch-free M padding
    v2f bf = *(const v2f*)(wrow + kb);
    // 8 args: (neg_a, A, neg_b, B, c_mod, C, reuse_a, reuse_b)
    acc = __builtin_amdgcn_wmma_f32_16x16x4_f32(
        false, a, false, bf, (short)0, acc, false, false);
  }

  // C layout: lane l, VGPR v -> C[v + 8*(l/16)][l%16]; rows 0..3 live in lanes 0..15, v=0..3
  if (lane < 16) {
    const int n = n0 + lane;
    const float bv = bias[n];
    #pragma unroll
    for (int v = 0; v < NCTX_; ++v) {
      out[obase + (long long)v * VD_ + n] = acc[v] + bv;
    }
  }
}

// ---------------------------------------------------------------------------
extern "C" void kernel_launch(void* const* d_in, const int* in_sizes, int n_in,
                              void* d_out, int out_size, void* d_ws, size_t ws_size,
                              hipStream_t stream) {
  const int*   text   = (const int*)  d_in[0];
  const float* emb    = (const float*)d_in[1];
  const float* ctx    = (const float*)d_in[2];
  const float* proj_w = (const float*)d_in[3];
  const float* proj_b = (const float*)d_in[4];
  const float* cpt    = (const float*)d_in[5];
  const float* cw     = (const float*)d_in[6];
  const float* cb     = (const float*)d_in[7];
  float* out = (float*)d_out;
  int* tok_ws = (int*)d_ws;   // B_*L_ ints = 1.26 MB scratch

  // 1) prompt tokens (+ int token scratch for the gather)
  build_pt_kernel<<<B_, 32, 0, stream>>>(text, out, tok_ws);

  // 2) the big embedding gather (memory-bound: ~1.3 GB traffic -> ~56 us roofline)
  const long long totalF4 = (long long)B_ * L_ * (D_ / 4);
  const int gblocks = (int)((totalF4 + 255) / 256);
  gather_pe_kernel<<<gblocks, 256, 0, stream>>>(emb, ctx, tok_ws, out + OFF_PE);

  // 3) ctx / cpt pass-through copies
  const int csize = (DEPTH_ - 1) * NCTX_ * D_;
  copy_small_kernel<<<(csize + 255) / 256, 256, 0, stream>>>(ctx, cpt, out);

  // 4) WMMA GEMMs: proj_ctx + vdp (9 slices * 48 N-tiles)
  wmma_proj_kernel<<<(VD_ / 16) * DEPTH_, 32, 0, stream>>>(
      ctx, proj_w, proj_b, cpt, cw, cb, out);
}